// STAN_57561151701209
// MI455X (gfx1250) — compile-verified
//
#include <hip/hip_runtime.h>
#include <math.h>

#define NN    10000   // nodes
#define TT    8       // timesteps
#define FIN   4
#define HEADS 4
#define C1    16      // per-head channels layer 1
#define HD2   32      // layer-2 channels
#define GG    32      // GRU hidden
#define PWn   16

typedef float v2f __attribute__((ext_vector_type(2)));
typedef float v8f __attribute__((ext_vector_type(8)));

// ---------------------------------------------------------------- helpers
__device__ __forceinline__ float lrelu02(float x) { return x > 0.f ? x : 0.2f * x; }
__device__ __forceinline__ float eluf(float x)    { return x > 0.f ? x : (__expf(x) - 1.f); }

__device__ __forceinline__ void atomicMaxF(float* addr, float v) {
    // signed/unsigned int ordering trick; addr initialized to -inf (0xFF800000)
    if (v >= 0.f) atomicMax((int*)addr, __float_as_int(v));
    else          atomicMin((unsigned int*)addr, __float_as_uint(v));
}

__device__ __forceinline__ void edge_sd(const int* __restrict__ ei, int E, int e,
                                        int& s, int& d) {
    if (e < E) { s = ei[e]; d = ei[E + e]; }
    else       { s = e - E; d = s; }          // self loops appended
}

__global__ void k_fill(float* __restrict__ p, float v, int n) {
    int i = blockIdx.x * 256 + threadIdx.x;
    if (i < n) p[i] = v;
}

// ---------------------------------------------------------------- GEMM 1 (WMMA)
// z1[t*NN+n][64] = x[t][n][0:4] @ W1[4][64]   (M=80000, K=4, N=64)
__global__ __launch_bounds__(32)
void k_gemm1(const float* __restrict__ dyn, const float* __restrict__ W1,
             float* __restrict__ z1) {
    const int tile = blockIdx.x;           // 5000 tiles of 16 rows
    const int lane = threadIdx.x;
    const int mlo  = lane & 15;
    const int hi   = lane >> 4;
    const int row  = tile * 16 + mlo;      // flattened t*NN+n
    const int t    = row / NN;
    const int n    = row - t * NN;
    // A fragment: a[j] = x[row][2*hi+j]; dynamic is [N, T, F]
    const float* ap = dyn + ((size_t)n * TT + t) * FIN + 2 * hi;
    v2f a; a[0] = ap[0]; a[1] = ap[1];
#pragma unroll
    for (int ct = 0; ct < 4; ++ct) {
        const int col = ct * 16 + mlo;
        v2f b;
        b[0] = W1[(2 * hi + 0) * 64 + col];
        b[1] = W1[(2 * hi + 1) * 64 + col];
        v8f c = {};
        c = __builtin_amdgcn_wmma_f32_16x16x4_f32(false, a, false, b, (short)0, c,
                                                  false, false);
#pragma unroll
        for (int r = 0; r < 8; ++r) {
            int orow = tile * 16 + r + 8 * hi;
            z1[(size_t)orow * 64 + col] = c[r];
        }
    }
}

// ---------------------------------------------------------------- GEMM 2 (WMMA)
// z2[row][32] = h1[row][0:64] @ W2[64][32]   (M=80000, K=64, N=32)
__global__ __launch_bounds__(32)
void k_gemm2(const float* __restrict__ h1, const float* __restrict__ W2,
             float* __restrict__ z2) {
    const int tile = blockIdx.x;
    const int lane = threadIdx.x;
    const int mlo  = lane & 15;
    const int hi   = lane >> 4;
    const int row  = tile * 16 + mlo;
    v8f acc0 = {}, acc1 = {};
#pragma unroll
    for (int kk = 0; kk < 16; ++kk) {
        const float* ap = h1 + (size_t)row * 64 + kk * 4 + 2 * hi;
        v2f a; a[0] = ap[0]; a[1] = ap[1];
        v2f b0, b1;
        b0[0] = W2[(kk * 4 + 2 * hi + 0) * 32 + mlo];
        b0[1] = W2[(kk * 4 + 2 * hi + 1) * 32 + mlo];
        b1[0] = W2[(kk * 4 + 2 * hi + 0) * 32 + 16 + mlo];
        b1[1] = W2[(kk * 4 + 2 * hi + 1) * 32 + 16 + mlo];
        acc0 = __builtin_amdgcn_wmma_f32_16x16x4_f32(false, a, false, b0, (short)0,
                                                     acc0, false, false);
        acc1 = __builtin_amdgcn_wmma_f32_16x16x4_f32(false, a, false, b1, (short)0,
                                                     acc1, false, false);
    }
#pragma unroll
    for (int r = 0; r < 8; ++r) {
        int orow = tile * 16 + r + 8 * hi;
        z2[(size_t)orow * 32 + mlo]      = acc0[r];
        z2[(size_t)orow * 32 + 16 + mlo] = acc1[r];
    }
}

// ---------------------------------------------------------------- attention dots
__global__ void k_att1(const float* __restrict__ z1, const float* __restrict__ as,
                       const float* __restrict__ ad, float* __restrict__ asrc,
                       float* __restrict__ adst) {
    int idx = blockIdx.x * 256 + threadIdx.x;      // over 80000*4
    if (idx >= TT * NN * HEADS) return;
    int h = idx & 3, r = idx >> 2;
    const float* zp = z1 + (size_t)r * 64 + h * 16;
    float sa = 0.f, sd = 0.f;
#pragma unroll
    for (int c = 0; c < 16; ++c) { float zv = zp[c]; sa += zv * as[h * 16 + c]; sd += zv * ad[h * 16 + c]; }
    asrc[idx] = sa; adst[idx] = sd;
}

__global__ void k_att2(const float* __restrict__ z2, const float* __restrict__ as,
                       const float* __restrict__ ad, float* __restrict__ asrc,
                       float* __restrict__ adst) {
    int r = blockIdx.x * 256 + threadIdx.x;        // over 80000
    if (r >= TT * NN) return;
    const float* zp = z2 + (size_t)r * 32;
    float sa = 0.f, sd = 0.f;
#pragma unroll
    for (int c = 0; c < 32; ++c) { float zv = zp[c]; sa += zv * as[c]; sd += zv * ad[c]; }
    asrc[r] = sa; adst[r] = sd;
}

// ---------------------------------------------------------------- GAT1 softmax
__global__ void k_emax1(const int* __restrict__ ei, int E, int Etot,
                        const float* __restrict__ asrc, const float* __restrict__ adst,
                        float* __restrict__ m) {
    int idx = blockIdx.x * 256 + threadIdx.x;      // Etot*T*H, T*H=32
    if (idx >= Etot * 32) return;
    int h = idx & 3, t = (idx >> 2) & 7, e = idx >> 5;
    int s, d; edge_sd(ei, E, e, s, d);
    float ec = lrelu02(asrc[(t * NN + s) * 4 + h] + adst[(t * NN + d) * 4 + h]);
    atomicMaxF(&m[(t * NN + d) * 4 + h], ec);
}

__global__ void k_esum1(const int* __restrict__ ei, int E, int Etot,
                        const float* __restrict__ asrc, const float* __restrict__ adst,
                        const float* __restrict__ m, float* __restrict__ sarr) {
    int idx = blockIdx.x * 256 + threadIdx.x;
    if (idx >= Etot * 32) return;
    int h = idx & 3, t = (idx >> 2) & 7, e = idx >> 5;
    int s, d; edge_sd(ei, E, e, s, d);
    int di = (t * NN + d) * 4 + h;
    float ec = lrelu02(asrc[(t * NN + s) * 4 + h] + adst[di]);
    atomicAdd(&sarr[di], __expf(ec - m[di]));
}

__global__ void k_agg1(const int* __restrict__ ei, int E, int Etot,
                       const float* __restrict__ asrc, const float* __restrict__ adst,
                       const float* __restrict__ m, const float* __restrict__ sarr,
                       const float* __restrict__ z1, float* __restrict__ g1) {
    int idx = blockIdx.x * 256 + threadIdx.x;
    if (idx >= Etot * 32) return;
    int h = idx & 3, t = (idx >> 2) & 7, e = idx >> 5;
    int s, d; edge_sd(ei, E, e, s, d);
    int di = (t * NN + d) * 4 + h;
    float ec  = lrelu02(asrc[(t * NN + s) * 4 + h] + adst[di]);
    float att = __expf(ec - m[di]) / sarr[di];
    const float* zp = z1 + ((size_t)(t * NN + s) * 64 + h * 16);
    float*       op = g1 + ((size_t)(t * NN + d) * 64 + h * 16);
#pragma unroll
    for (int c = 0; c < 16; ++c) atomicAdd(&op[c], zp[c] * att);
}

__global__ void k_elu1(float* __restrict__ g, const float* __restrict__ b1) {
    int idx = blockIdx.x * 256 + threadIdx.x;      // over 80000*64
    if (idx >= TT * NN * 64) return;
    g[idx] = eluf(g[idx] + b1[idx & 63]);
}

// ---------------------------------------------------------------- GAT2 softmax (H=1)
__global__ void k_emax2(const int* __restrict__ ei, int E, int Etot,
                        const float* __restrict__ asrc, const float* __restrict__ adst,
                        float* __restrict__ m) {
    int idx = blockIdx.x * 256 + threadIdx.x;      // Etot*T
    if (idx >= Etot * 8) return;
    int t = idx & 7, e = idx >> 3;
    int s, d; edge_sd(ei, E, e, s, d);
    float ec = lrelu02(asrc[t * NN + s] + adst[t * NN + d]);
    atomicMaxF(&m[t * NN + d], ec);
}

__global__ void k_esum2(const int* __restrict__ ei, int E, int Etot,
                        const float* __restrict__ asrc, const float* __restrict__ adst,
                        const float* __restrict__ m, float* __restrict__ sarr) {
    int idx = blockIdx.x * 256 + threadIdx.x;
    if (idx >= Etot * 8) return;
    int t = idx & 7, e = idx >> 3;
    int s, d; edge_sd(ei, E, e, s, d);
    int di = t * NN + d;
    float ec = lrelu02(asrc[t * NN + s] + adst[di]);
    atomicAdd(&sarr[di], __expf(ec - m[di]));
}

__global__ void k_agg2(const int* __restrict__ ei, int E, int Etot,
                       const float* __restrict__ asrc, const float* __restrict__ adst,
                       const float* __restrict__ m, const float* __restrict__ sarr,
                       const float* __restrict__ z2, float* __restrict__ g2) {
    int idx = blockIdx.x * 256 + threadIdx.x;
    if (idx >= Etot * 8) return;
    int t = idx & 7, e = idx >> 3;
    int s, d; edge_sd(ei, E, e, s, d);
    int di = t * NN + d;
    float ec  = lrelu02(asrc[t * NN + s] + adst[di]);
    float att = __expf(ec - m[di]) / sarr[di];
    const float* zp = z2 + (size_t)(t * NN + s) * 32;
    float*       op = g2 + (size_t)di * 32;
#pragma unroll
    for (int c = 0; c < 32; ++c) atomicAdd(&op[c], zp[c] * att);
}

// ELU(+b2) and mean over nodes -> curh[T][32] (raw sum, tail divides by NN)
__global__ __launch_bounds__(256)
void k_red2(const float* __restrict__ g2, const float* __restrict__ b2,
            float* __restrict__ curh) {
    __shared__ float sm[256];
    int blk = blockIdx.x;                 // 10000 blocks = T * 1250
    int t = blk / 1250;
    int nbase = (blk % 1250) * 8;
    int tid = threadIdx.x;
    int c = tid & 31, ro = tid >> 5;
    int n = nbase + ro;
    float v = eluf(g2[((size_t)(t * NN + n)) * 32 + c] + b2[c]);
    sm[tid] = v;
    __syncthreads();
    for (int st = 128; st >= 32; st >>= 1) {
        if (tid < st) sm[tid] += sm[tid + st];
        __syncthreads();
    }
    if (tid < 32) atomicAdd(&curh[t * 32 + tid], sm[tid]);
}

// ---------------------------------------------------------------- tail: GRU + heads + SIR
__global__ __launch_bounds__(32)
void k_tail(const float* __restrict__ curh, const float* __restrict__ cI,
            const float* __restrict__ cR, const float* __restrict__ Nv,
            const float* __restrict__ Iv, const float* __restrict__ Rv,
            const float* __restrict__ h0, const float* __restrict__ W_ih,
            const float* __restrict__ W_hh, const float* __restrict__ b_ih,
            const float* __restrict__ b_hh, const float* __restrict__ Wi,
            const float* __restrict__ bi, const float* __restrict__ Wr,
            const float* __restrict__ br, const float* __restrict__ Ws,
            const float* __restrict__ bs, float* __restrict__ out) {
    __shared__ float h[GG], xt[HD2], hs[TT][GG], alpha[TT], beta[TT];
    const int g = threadIdx.x;            // 0..31
    h[g] = h0[g];
    __syncthreads();
    for (int t = 0; t < TT; ++t) {
        xt[g] = curh[t * 32 + g] * (1.0f / NN);
        __syncthreads();
        float gr = b_ih[g], gz = b_ih[32 + g], gn = b_ih[64 + g];
        float hr = b_hh[g], hz = b_hh[32 + g], hn_ = b_hh[64 + g];
#pragma unroll
        for (int k = 0; k < 32; ++k) {
            float xk = xt[k], hk = h[k];
            gr  += W_ih[g * 32 + k] * xk;
            gz  += W_ih[(32 + g) * 32 + k] * xk;
            gn  += W_ih[(64 + g) * 32 + k] * xk;
            hr  += W_hh[g * 32 + k] * hk;
            hz  += W_hh[(32 + g) * 32 + k] * hk;
            hn_ += W_hh[(64 + g) * 32 + k] * hk;
        }
        float r  = 1.f / (1.f + __expf(-(gr + hr)));
        float zg = 1.f / (1.f + __expf(-(gz + hz)));
        float ng = tanhf(gn + r * hn_);
        float hnew = (1.f - zg) * ng + zg * h[g];
        __syncthreads();
        h[g] = hnew; hs[t][g] = hnew;
        __syncthreads();
    }
    out[512 + g] = h[g];                  // h_final
    for (int t = 0; t < TT; ++t) {
        if (g < PWn) {
            float aI = bi[g], aR = br[g];
            for (int j = 0; j < 32; ++j) { aI += Wi[g * 34 + j] * hs[t][j]; aR += Wr[g * 34 + j] * hs[t][j]; }
            aI += Wi[g * 34 + 32] * cI[t] + Wi[g * 34 + 33] * cR[t];
            aR += Wr[g * 34 + 32] * cI[t] + Wr[g * 34 + 33] * cR[t];
            out[t * PWn + g]       = aI;  // new_I
            out[128 + t * PWn + g] = aR;  // new_R
        }
        if (g < 2) {
            float acc = bs[g];
            for (int j = 0; j < 32; ++j) acc += Ws[g * 34 + j] * hs[t][j];
            acc += Ws[g * 34 + 32] * cI[t] + Ws[g * 34 + 33] * cR[t];
            float sg = 1.f / (1.f + __expf(-acc));
            if (g == 0) alpha[t] = sg; else beta[t] = sg;
        }
    }
    __syncthreads();
    if (g < TT) {
        int t = g;
        float Np = Nv[0];
        float lI = Iv[t], lR = Rv[t];
        for (int p = 0; p < PWn; ++p) {
            float lS = Np - lI - lR;
            float dI = alpha[t] * lI * (lS / Np) - beta[t] * lI;
            float dR = beta[t] * lI;
            out[256 + t * PWn + p] = dI;  // phy_I
            out[384 + t * PWn + p] = dR;  // phy_R
            lI += dI; lR += dR;
        }
    }
}

// ---------------------------------------------------------------- launcher
extern "C" void kernel_launch(void* const* d_in, const int* in_sizes, int n_in,
                              void* d_out, int out_size, void* d_ws, size_t ws_size,
                              hipStream_t stream) {
    (void)n_in; (void)out_size; (void)ws_size;
    const float* dyn  = (const float*)d_in[0];
    const float* cI   = (const float*)d_in[1];
    const float* cR   = (const float*)d_in[2];
    const float* Nv   = (const float*)d_in[3];
    const float* Iv   = (const float*)d_in[4];
    const float* Rv   = (const float*)d_in[5];
    const int*   ei   = (const int*)d_in[6];
    const float* h0   = (const float*)d_in[7];
    const float* W1   = (const float*)d_in[8];
    const float* as1  = (const float*)d_in[9];
    const float* ad1  = (const float*)d_in[10];
    const float* b1   = (const float*)d_in[11];
    const float* W2   = (const float*)d_in[12];
    const float* as2  = (const float*)d_in[13];
    const float* ad2  = (const float*)d_in[14];
    const float* b2   = (const float*)d_in[15];
    const float* W_ih = (const float*)d_in[16];
    const float* W_hh = (const float*)d_in[17];
    const float* b_ih = (const float*)d_in[18];
    const float* b_hh = (const float*)d_in[19];
    const float* Wi   = (const float*)d_in[20];
    const float* bi   = (const float*)d_in[21];
    const float* Wr   = (const float*)d_in[22];
    const float* br   = (const float*)d_in[23];
    const float* Ws   = (const float*)d_in[24];
    const float* bs   = (const float*)d_in[25];
    float* out = (float*)d_out;

    const int E    = in_sizes[6] / 2;     // 160000
    const int Etot = E + NN;              // + self loops

    // workspace carve-up (floats)
    float* p = (float*)d_ws;
    float* z1    = p; p += (size_t)TT * NN * 64;   // 5.12M
    float* asrc1 = p; p += TT * NN * HEADS;        // 320k
    float* adst1 = p; p += TT * NN * HEADS;
    float* m1    = p; p += TT * NN * HEADS;
    float* s1    = p; p += TT * NN * HEADS;
    float* g1    = p; p += (size_t)TT * NN * 64;   // 5.12M (becomes h1 post-ELU)
    float* z2    = p; p += (size_t)TT * NN * 32;   // 2.56M
    float* asrc2 = p; p += TT * NN;
    float* adst2 = p; p += TT * NN;
    float* m2    = p; p += TT * NN;
    float* s2    = p; p += TT * NN;
    float* g2    = p; p += (size_t)TT * NN * 32;   // 2.56M
    float* curh  = p; p += 256;

    auto nb = [](long n) { return (unsigned)((n + 255) / 256); };
    const float NEGINF = -__builtin_huge_valf();

    // per-call init (deterministic; d_ws is not re-poisoned between replays)
    k_fill<<<nb(TT * NN * HEADS), 256, 0, stream>>>(m1, NEGINF, TT * NN * HEADS);
    k_fill<<<nb(TT * NN * HEADS), 256, 0, stream>>>(s1, 0.f, TT * NN * HEADS);
    k_fill<<<nb((long)TT * NN * 64), 256, 0, stream>>>(g1, 0.f, TT * NN * 64);
    k_fill<<<nb(TT * NN), 256, 0, stream>>>(m2, NEGINF, TT * NN);
    k_fill<<<nb(TT * NN), 256, 0, stream>>>(s2, 0.f, TT * NN);
    k_fill<<<nb((long)TT * NN * 32), 256, 0, stream>>>(g2, 0.f, TT * NN * 32);
    k_fill<<<1, 256, 0, stream>>>(curh, 0.f, 256);

    // GAT layer 1
    k_gemm1<<<(TT * NN) / 16, 32, 0, stream>>>(dyn, W1, z1);
    k_att1<<<nb(TT * NN * HEADS), 256, 0, stream>>>(z1, as1, ad1, asrc1, adst1);
    k_emax1<<<nb((long)Etot * 32), 256, 0, stream>>>(ei, E, Etot, asrc1, adst1, m1);
    k_esum1<<<nb((long)Etot * 32), 256, 0, stream>>>(ei, E, Etot, asrc1, adst1, m1, s1);
    k_agg1<<<nb((long)Etot * 32), 256, 0, stream>>>(ei, E, Etot, asrc1, adst1, m1, s1, z1, g1);
    k_elu1<<<nb((long)TT * NN * 64), 256, 0, stream>>>(g1, b1);

    // GAT layer 2
    k_gemm2<<<(TT * NN) / 16, 32, 0, stream>>>(g1, W2, z2);
    k_att2<<<nb(TT * NN), 256, 0, stream>>>(z2, as2, ad2, asrc2, adst2);
    k_emax2<<<nb((long)Etot * 8), 256, 0, stream>>>(ei, E, Etot, asrc2, adst2, m2);
    k_esum2<<<nb((long)Etot * 8), 256, 0, stream>>>(ei, E, Etot, asrc2, adst2, m2, s2);
    k_agg2<<<nb((long)Etot * 8), 256, 0, stream>>>(ei, E, Etot, asrc2, adst2, m2, s2, z2, g2);
    k_red2<<<TT * 1250, 256, 0, stream>>>(g2, b2, curh);

    // GRU + heads + SIR rollout
    k_tail<<<1, 32, 0, stream>>>(curh, cI, cR, Nv, Iv, Rv, h0, W_ih, W_hh, b_ih, b_hh,
                                 Wi, bi, Wr, br, Ws, bs, out);
}